// FlexibleGATNet_9972914061865
// MI455X (gfx1250) — compile-verified
//
#include <hip/hip_runtime.h>
#include <hip/hip_bf16.h>

// ---------------- problem constants (match reference) ----------------
#define NN   20000
#define EE   320000
#define GG   64
#define IND  6
#define EDD  2
#define HH   4
#define FFD  64
#define HF   256   // HH*FFD
#define FCD  64
#define NCD  2
#define E2   (EE + NN)   // 340000 edges after self loops
#define KK   256         // GEMM reduction dim (always HF here)

typedef __attribute__((ext_vector_type(2))) float v2f;
typedef __attribute__((ext_vector_type(8))) float v8f;

// ---------------- small device helpers ----------------
__device__ __forceinline__ float d_elu(float v) {
    return v > 0.f ? v : (expf(v) - 1.f);
}
// monotonic float<->uint encoding for atomic max on f32
__device__ __forceinline__ unsigned f2ord(float f) {
    unsigned u = __float_as_uint(f);
    return (u & 0x80000000u) ? ~u : (u | 0x80000000u);
}
__device__ __forceinline__ float ord2f(unsigned o) {
    return __uint_as_float((o & 0x80000000u) ? (o ^ 0x80000000u) : ~o);
}

// ---------------- generic zero fill ----------------
__global__ void k_zero_u32(unsigned* p, int n) {
    int i = blockIdx.x * blockDim.x + threadIdx.x;
    if (i < n) p[i] = 0u;
}

// ---------------- self-loop mean edge attr ----------------
__global__ void k_selfloop_accum(const int* __restrict__ ei,
                                 const float* __restrict__ eattr,
                                 float* __restrict__ loop_attr,
                                 float* __restrict__ cnt) {
    int e = blockIdx.x * blockDim.x + threadIdx.x;
    if (e >= EE) return;
    int d = ei[EE + e];
    atomicAdd(&loop_attr[2 * d + 0], eattr[2 * e + 0]);
    atomicAdd(&loop_attr[2 * d + 1], eattr[2 * e + 1]);
    atomicAdd(&cnt[d], 1.0f);
}
__global__ void k_selfloop_final(float* __restrict__ loop_attr,
                                 const float* __restrict__ cnt) {
    int n = blockIdx.x * blockDim.x + threadIdx.x;
    if (n >= NN) return;
    float c = cnt[n]; c = c > 1.0f ? c : 1.0f;
    loop_attr[2 * n + 0] /= c;
    loop_attr[2 * n + 1] /= c;
}

// ---------------- layer-0 node transforms (K=6: VALU, fused 3 mats) ----------------
__global__ void k_proj0(const float* __restrict__ x,
                        const float* __restrict__ Wl, const float* __restrict__ bl,
                        const float* __restrict__ Wr, const float* __restrict__ br,
                        const float* __restrict__ Wp,
                        float* __restrict__ xl, float* __restrict__ xr,
                        float* __restrict__ xres) {
    int n = blockIdx.x;           // one node per block
    int j = threadIdx.x;          // 256 output features
    float xv[IND];
#pragma unroll
    for (int i = 0; i < IND; i++) xv[i] = x[n * IND + i];
    float sl = bl[j], sr = br[j], sp = 0.f;
#pragma unroll
    for (int i = 0; i < IND; i++) {
        sl += xv[i] * Wl[i * HF + j];
        sr += xv[i] * Wr[i * HF + j];
        sp += xv[i] * Wp[i * HF + j];
    }
    size_t o = (size_t)n * HF + j;
    xl[o] = sl; xr[o] = sr; xres[o] = sp;
}

// ---------------- edge logits: one wave32 per edge, b128 gathers ----------------
__device__ __forceinline__ float dot_lrelu4(float4 l, float4 r, float4 w0, float4 w1,
                                            float4 at, float ea0, float ea1) {
    float m, acc = 0.f;
    m = l.x + r.x + ea0 * w0.x + ea1 * w1.x; m = m > 0.f ? m : 0.2f * m; acc += m * at.x;
    m = l.y + r.y + ea0 * w0.y + ea1 * w1.y; m = m > 0.f ? m : 0.2f * m; acc += m * at.y;
    m = l.z + r.z + ea0 * w0.z + ea1 * w1.z; m = m > 0.f ? m : 0.2f * m; acc += m * at.z;
    m = l.w + r.w + ea0 * w0.w + ea1 * w1.w; m = m > 0.f ? m : 0.2f * m; acc += m * at.w;
    return acc;
}

__global__ void k_edge_logits(const int* __restrict__ ei,
                              const float* __restrict__ eattr,
                              const float* __restrict__ loop_attr,
                              const float* __restrict__ xl,
                              const float* __restrict__ xr,
                              const float* __restrict__ We,   // [2, HF]
                              const float* __restrict__ att,  // [H, F] == flat [HF]
                              float* __restrict__ exbuf,      // [E2, H] (raw logits)
                              unsigned* __restrict__ nodemax) // [N, H] ord-encoded
{
    int w = blockIdx.x * (blockDim.x >> 5) + (threadIdx.x >> 5);
    int lane = threadIdx.x & 31;
    if (w >= E2) return;   // uniform per wave
    int s, d; float ea0, ea1;
    if (w < EE) {
        s = ei[w]; d = ei[EE + w];
        ea0 = eattr[2 * w]; ea1 = eattr[2 * w + 1];
    } else {
        s = w - EE; d = s;
        ea0 = loop_attr[2 * s]; ea1 = loop_attr[2 * s + 1];
    }
    int f0 = lane * 8;     // 8 contiguous f per lane, head = lane>>3
    const float4* pl  = (const float4*)(xl + (size_t)s * HF + f0);
    const float4* pr  = (const float4*)(xr + (size_t)d * HF + f0);
    const float4* pw0 = (const float4*)(We + f0);
    const float4* pw1 = (const float4*)(We + HF + f0);
    const float4* pa  = (const float4*)(att + f0);
    float acc = dot_lrelu4(pl[0], pr[0], pw0[0], pw1[0], pa[0], ea0, ea1)
              + dot_lrelu4(pl[1], pr[1], pw0[1], pw1[1], pa[1], ea0, ea1);
    // reduce across the 8 lanes of this head
    acc += __shfl_xor(acc, 4, 32);
    acc += __shfl_xor(acc, 2, 32);
    acc += __shfl_xor(acc, 1, 32);
    if ((lane & 7) == 0) {
        int h = lane >> 3;
        exbuf[(size_t)w * HH + h] = acc;
        atomicMax(&nodemax[d * HH + h], f2ord(acc));
    }
}

// ---------------- softmax: exp(logit - max) and segment sum ----------------
__global__ void k_exp(const int* __restrict__ ei,
                      float* __restrict__ exbuf,
                      const unsigned* __restrict__ nodemax,
                      float* __restrict__ nodesum) {
    int idx = blockIdx.x * blockDim.x + threadIdx.x;
    if (idx >= E2 * HH) return;
    int e = idx >> 2, h = idx & 3;
    int d = (e < EE) ? ei[EE + e] : (e - EE);
    float mx = ord2f(nodemax[d * HH + h]);
    float v = expf(exbuf[idx] - mx);
    exbuf[idx] = v;
    atomicAdd(&nodesum[d * HH + h], v);
}

// ---------------- alpha-weighted scatter aggregation: wave per edge ----------------
__global__ void k_aggregate(const int* __restrict__ ei,
                            const float* __restrict__ exbuf,
                            const float* __restrict__ nodesum,
                            const float* __restrict__ xl,
                            float* __restrict__ hout) {
    int w = blockIdx.x * (blockDim.x >> 5) + (threadIdx.x >> 5);
    int lane = threadIdx.x & 31;
    if (w >= E2) return;
    int s, d;
    if (w < EE) { s = ei[w]; d = ei[EE + w]; }
    else        { s = w - EE; d = s; }
    int f0 = lane * 8, h = lane >> 3;
    float alpha = exbuf[(size_t)w * HH + h] / nodesum[d * HH + h];
    const float4* pl = (const float4*)(xl + (size_t)s * HF + f0);
    float4 l0 = pl[0], l1 = pl[1];
    float* po = hout + (size_t)d * HF + f0;
    atomicAdd(&po[0], alpha * l0.x);
    atomicAdd(&po[1], alpha * l0.y);
    atomicAdd(&po[2], alpha * l0.z);
    atomicAdd(&po[3], alpha * l0.w);
    atomicAdd(&po[4], alpha * l1.x);
    atomicAdd(&po[5], alpha * l1.y);
    atomicAdd(&po[6], alpha * l1.z);
    atomicAdd(&po[7], alpha * l1.w);
}

// ---------------- epilogue: elu(h + bias) + residual (float4) ----------------
__global__ void k_epilogue(const float* __restrict__ hbuf,
                           const float* __restrict__ bias,
                           const float* __restrict__ res,
                           float* __restrict__ outp) {
    int idx = blockIdx.x * blockDim.x + threadIdx.x;
    if (idx >= NN * HF / 4) return;
    int j4 = (idx * 4) & (HF - 1);
    float4 hv = ((const float4*)hbuf)[idx];
    float4 rv = ((const float4*)res)[idx];
    float4 bv = *(const float4*)(bias + j4);
    float4 o;
    o.x = d_elu(hv.x + bv.x) + rv.x;
    o.y = d_elu(hv.y + bv.y) + rv.y;
    o.z = d_elu(hv.z + bv.z) + rv.z;
    o.w = d_elu(hv.w + bv.w) + rv.w;
    ((float4*)outp)[idx] = o;
}

// ---------------- WMMA fp32 GEMM: C = A[M,KK] @ W[KK,Ncol] + bias (+opt ELU) ----
// Block = 8 waves; all waves share one 256x16 W column-block staged in LDS (16 KB).
// Each wave computes one 16x16 C tile with V_WMMA_F32_16X16X4_F32, K stepped by 4.
// A fragment per K-step comes from a single b128 load (float4) per lane; the lane
// half (K 0/1 vs 2/3) is selected with v_cndmask so EXEC stays all-ones for WMMA.
__global__ void k_gemm_wmma(const float* __restrict__ A,
                            const float* __restrict__ W,
                            const float* __restrict__ bias,
                            float* __restrict__ C,
                            int M, int Ncol, int act) {
    __shared__ float Bs[KK * 16];
    int tid = threadIdx.x;                 // 256 threads
    int n0 = blockIdx.y * 16;
    {   // stage W[:, n0:n0+16] into LDS, coalesced 16-wide rows
        int col = tid & 15, rbase = tid >> 4;
        for (int r = rbase; r < KK; r += 16)
            Bs[r * 16 + col] = W[(size_t)r * Ncol + n0 + col];
    }
    __syncthreads();
    int wid = tid >> 5, lane = tid & 31;
    int rt = blockIdx.x * 8 + wid;         // row tile (M assumed multiple of 16)
    if (rt * 16 >= M) return;              // wave-uniform exit
    int half = lane >> 4, lm = lane & 15;
    int m0 = rt * 16;
    const float* Arow = A + (size_t)(m0 + lm) * KK;
    v8f acc = {};
    for (int k = 0; k < KK; k += 4) {
        float4 av = *(const float4*)(Arow + k);   // one b128 per lane per K-step
        int ka = k + half * 2;
        v2f a, b;
        a.x = half ? av.z : av.x;
        a.y = half ? av.w : av.y;
        b.x = Bs[ka * 16 + lm];
        b.y = Bs[(ka + 1) * 16 + lm];
        acc = __builtin_amdgcn_wmma_f32_16x16x4_f32(
                  false, a, false, b, (short)0, acc, false, false);
    }
    float bj = bias[n0 + lm];
#pragma unroll
    for (int v = 0; v < 8; v++) {
        float val = acc[v] + bj;
        if (act) val = d_elu(val);
        C[(size_t)(m0 + v + half * 8) * Ncol + n0 + lm] = val;
    }
}

// ---------------- mean pool + classifier ----------------
__global__ void k_pool(const float* __restrict__ z, const int* __restrict__ batch,
                       float* __restrict__ pooled, float* __restrict__ gcnt) {
    int n = blockIdx.x;            // node per block, one feature per thread (64 threads)
    int j = threadIdx.x;
    int g = batch[n];
    atomicAdd(&pooled[g * FCD + j], z[(size_t)n * FCD + j]);
    if (j == 0) atomicAdd(&gcnt[g], 1.0f);
}
__global__ void k_classify(const float* __restrict__ pooled,
                           const float* __restrict__ gcnt,
                           const float* __restrict__ Wc,
                           const float* __restrict__ bc,
                           float* __restrict__ out) {
    int idx = threadIdx.x;
    if (idx >= GG * NCD) return;
    int g = idx / NCD, c = idx % NCD;
    float cn = gcnt[g]; cn = cn > 1.f ? cn : 1.f;
    float s = bc[c];
    for (int k = 0; k < FCD; k++)
        s += (pooled[g * FCD + k] / cn) * Wc[k * NCD + c];
    out[idx] = s;
}

// ---------------- workspace layout (float offsets; all 16B-aligned) ----------------
static const size_t OFF_LOOP = 0;              // [N*2]
static const size_t OFF_CNT  = 40000;          // [N]
static const size_t OFF_NMAX = 60000;          // [N*H] (uint)
static const size_t OFF_NSUM = 140000;         // [N*H]
static const size_t OFF_EX   = 220000;         // [E2*H]
static const size_t OFF_XL   = 1580000;        // [N*HF]
static const size_t OFF_XR   = 6700000;        // [N*HF]
static const size_t OFF_HOUT = 11820000;       // [N*HF]
static const size_t OFF_XRES = 16940000;       // [N*HF]
static const size_t OFF_X1   = 22060000;       // [N*HF]
static const size_t OFF_X2   = 27180000;       // [N*HF]
static const size_t OFF_Z    = 32300000;       // [N*FC]
static const size_t OFF_POOL = 33580000;       // [G*FC]
static const size_t OFF_GCNT = 33584096;       // [G]

static inline int ceil_div(int a, int b) { return (a + b - 1) / b; }

extern "C" void kernel_launch(void* const* d_in, const int* in_sizes, int n_in,
                              void* d_out, int out_size, void* d_ws, size_t ws_size,
                              hipStream_t stream) {
    const float* x     = (const float*)d_in[0];
    const int*   ei    = (const int*)  d_in[1];
    const float* eattr = (const float*)d_in[2];
    const int*   batch = (const int*)  d_in[3];
    const float* Wl0 = (const float*)d_in[4];   const float* bl0 = (const float*)d_in[5];
    const float* Wr0 = (const float*)d_in[6];   const float* br0 = (const float*)d_in[7];
    const float* We0 = (const float*)d_in[8];   const float* att0 = (const float*)d_in[9];
    const float* b0  = (const float*)d_in[10];  const float* proj0 = (const float*)d_in[11];
    const float* Wl1 = (const float*)d_in[12];  const float* bl1 = (const float*)d_in[13];
    const float* Wr1 = (const float*)d_in[14];  const float* br1 = (const float*)d_in[15];
    const float* We1 = (const float*)d_in[16];  const float* att1 = (const float*)d_in[17];
    const float* b1  = (const float*)d_in[18];
    const float* Wp  = (const float*)d_in[19];  const float* bp = (const float*)d_in[20];
    const float* Wc  = (const float*)d_in[21];  const float* bc = (const float*)d_in[22];
    float* out = (float*)d_out;

    float* ws = (float*)d_ws;
    float*    loop_attr = ws + OFF_LOOP;
    float*    cnt       = ws + OFF_CNT;
    unsigned* nodemax   = (unsigned*)(ws + OFF_NMAX);
    float*    nodesum   = ws + OFF_NSUM;
    float*    exbuf     = ws + OFF_EX;
    float*    xl        = ws + OFF_XL;
    float*    xr        = ws + OFF_XR;
    float*    hout      = ws + OFF_HOUT;
    float*    xres      = ws + OFF_XRES;
    float*    x1        = ws + OFF_X1;
    float*    x2        = ws + OFF_X2;
    float*    z         = ws + OFF_Z;
    float*    pooled    = ws + OFF_POOL;
    float*    gcnt      = ws + OFF_GCNT;

    const int edge_waves_grid = ceil_div(E2, 8);     // 8 waves (256 thr) per block
    const int gemm_rows_grid  = ceil_div(NN / 16, 8); // 8 row-tiles per block

    // 1) self-loop mean edge attrs
    k_zero_u32<<<ceil_div(60000, 256), 256, 0, stream>>>((unsigned*)loop_attr, 60000);
    k_selfloop_accum<<<ceil_div(EE, 256), 256, 0, stream>>>(ei, eattr, loop_attr, cnt);
    k_selfloop_final<<<ceil_div(NN, 256), 256, 0, stream>>>(loop_attr, cnt);

    // 2) layer-0 node transforms (K=6, VALU) + residual projection
    k_proj0<<<NN, HF, 0, stream>>>(x, Wl0, bl0, Wr0, br0, proj0, xl, xr, xres);

    // 3) layer-0 attention
    k_zero_u32<<<ceil_div(160000, 256), 256, 0, stream>>>(nodemax, 160000); // nodemax+nodesum
    k_zero_u32<<<ceil_div(NN * HF, 256), 256, 0, stream>>>((unsigned*)hout, NN * HF);
    k_edge_logits<<<edge_waves_grid, 256, 0, stream>>>(ei, eattr, loop_attr, xl, xr,
                                                       We0, att0, exbuf, nodemax);
    k_exp<<<ceil_div(E2 * HH, 256), 256, 0, stream>>>(ei, exbuf, nodemax, nodesum);
    k_aggregate<<<edge_waves_grid, 256, 0, stream>>>(ei, exbuf, nodesum, xl, hout);
    k_epilogue<<<ceil_div(NN * HF / 4, 256), 256, 0, stream>>>(hout, b0, xres, x1);

    // 4) layer-1 node transforms: fp32 WMMA GEMMs (20000x256 @ 256x256)
    {
        dim3 grid(gemm_rows_grid, HF / 16);
        k_gemm_wmma<<<grid, 256, 0, stream>>>(x1, Wl1, bl1, xl, NN, HF, 0);
        k_gemm_wmma<<<grid, 256, 0, stream>>>(x1, Wr1, br1, xr, NN, HF, 0);
    }

    // 5) layer-1 attention
    k_zero_u32<<<ceil_div(160000, 256), 256, 0, stream>>>(nodemax, 160000);
    k_zero_u32<<<ceil_div(NN * HF, 256), 256, 0, stream>>>((unsigned*)hout, NN * HF);
    k_edge_logits<<<edge_waves_grid, 256, 0, stream>>>(ei, eattr, loop_attr, xl, xr,
                                                       We1, att1, exbuf, nodemax);
    k_exp<<<ceil_div(E2 * HH, 256), 256, 0, stream>>>(ei, exbuf, nodemax, nodesum);
    k_aggregate<<<edge_waves_grid, 256, 0, stream>>>(ei, exbuf, nodesum, xl, hout);
    k_epilogue<<<ceil_div(NN * HF / 4, 256), 256, 0, stream>>>(hout, b1, x1, x2);

    // 6) head MLP (WMMA, fused ELU), pool, classify
    {
        dim3 grid(gemm_rows_grid, FCD / 16);
        k_gemm_wmma<<<grid, 256, 0, stream>>>(x2, Wp, bp, z, NN, FCD, 1);
    }
    k_zero_u32<<<ceil_div(GG * FCD + GG, 256), 256, 0, stream>>>((unsigned*)pooled,
                                                                GG * FCD + GG);
    k_pool<<<NN, FCD, 0, stream>>>(z, batch, pooled, gcnt);
    k_classify<<<1, 128, 0, stream>>>(pooled, gcnt, Wc, bc, out);
}